// SimpleRetention_82583631167650
// MI455X (gfx1250) — compile-verified
//
#include <hip/hip_runtime.h>
#include <math.h>

typedef float v2f __attribute__((ext_vector_type(2)));
typedef float v8f __attribute__((ext_vector_type(8)));
typedef unsigned int u32x4 __attribute__((ext_vector_type(4)));
typedef int i32x4 __attribute__((ext_vector_type(4)));
typedef int i32x8 __attribute__((ext_vector_type(8)));

#define S_LEN 4096
#define HD    64           // head dim == vdim
#define BM    64           // query rows per workgroup
#define NWAVE 4
#define LDSP  68           // padded LDS row stride (floats); TDM pad reproduces this
#define BUFSZ (BM * LDSP)  // one K or V buffer, in floats
#define LOG2_GAMMA (-0.045803689613824f)   // log2(0.96875)

#if defined(__has_builtin)
#if __has_builtin(__builtin_amdgcn_tensor_load_to_lds) && __has_builtin(__builtin_amdgcn_s_wait_tensorcnt)
#define HAVE_TDM 1
#endif
#endif
#ifndef HAVE_TDM
#define HAVE_TDM 0
#endif

// XPos rotation on an even/odd dim pair (2h, 2h+1) at position `pos`.
__device__ __forceinline__ void xpos_pair(float x0, float x1, int pos, int h, float sgn,
                                          float& o0, float& o1) {
    float fh = (float)h;
    float inv_freq = powf(10000.0f, -fh * (1.0f / 32.0f));     // half = 32
    float s, c;
    sincosf((float)pos * inv_freq, &s, &c);
    float base  = (2.0f * fh + 0.4f * 64.0f) * (1.0f / (1.4f * 64.0f));
    float scale = powf(base, sgn * (float)pos * (1.0f / 512.0f));
    s *= scale;
    c *= scale;
    o0 = x0 * c - x1 * s;
    o1 = x1 * c + x0 * s;
}

// One-shot preprocessing: Qh = XPos(Q, +scale), Kh = XPos(K, 1/scale).
__global__ __launch_bounds__(256, 1)
void xpos_rotate(const float* __restrict__ Q, const float* __restrict__ K,
                 float* __restrict__ Qh, float* __restrict__ Kh,
                 int total_pairs, int seqlen) {
    int idx = blockIdx.x * 256 + threadIdx.x;
    if (idx >= total_pairs) return;
    int h   = idx & 31;
    int pos = (idx >> 5) & (seqlen - 1);

    float fh = (float)h;
    float inv_freq = powf(10000.0f, -fh * (1.0f / 32.0f));
    float s, c;
    sincosf((float)pos * inv_freq, &s, &c);
    float base   = (2.0f * fh + 0.4f * 64.0f) * (1.0f / (1.4f * 64.0f));
    float scale  = powf(base, (float)pos * (1.0f / 512.0f));
    float rscale = 1.0f / scale;

    size_t off = (size_t)idx * 2;
    float q0 = Q[off], q1 = Q[off + 1];
    float k0 = K[off], k1 = K[off + 1];
    float sq = s * scale,  cq = c * scale;
    float sk = s * rscale, ck = c * rscale;
    Qh[off]     = q0 * cq - q1 * sq;
    Qh[off + 1] = q1 * cq + q0 * sq;
    Kh[off]     = k0 * ck - k1 * sk;
    Kh[off + 1] = k1 * ck + k0 * sk;
}

#if HAVE_TDM
// Issue one TDM load of a 64x64 fp32 tile (row-major, stride 64) into LDS at
// lds_byte_off, with hardware padding of 4 dwords after every 64 dwords
// (-> LDS row stride = 68 floats = LDSP). D# packing per CDNA5 ISA ch.8.
// This toolchain exposes the 6-arg builtin:
//   (uint32x4 g0, int32x8 g1, int32x4 g2, int32x4 g3, int32x8 extra, i32 cpol)
// groups 2/3 + extra are unused for a 2D tensor -> zero-filled.
__device__ __forceinline__ void tdm_load_tile(const float* gsrc, unsigned lds_byte_off,
                                              int rows_remaining) {
    unsigned long long ga = (unsigned long long)(uintptr_t)gsrc;
    u32x4 g0;
    g0[0] = 1u;                                          // count=1, user descriptor
    g0[1] = lds_byte_off;                                // lds_addr[31:0]
    g0[2] = (unsigned)ga;                                // global_addr[31:0]
    g0[3] = (unsigned)((ga >> 32) & 0x1FFFFFFu) | (2u << 30);  // addr[56:32], type=2
    i32x8 g1;
    g1[0] = (2 << 16) | (1 << 20) | (5 << 22) | (3 << 25); // 4B elems, pad_en,
                                                           // pad every 64dw, pad 4dw
    g1[1] = (HD & 0xFFFF) << 16;                         // abar=0; tensor_dim0[15:0]=64
    g1[2] = ((rows_remaining & 0xFFFF) << 16);           // dim0[31:16]=0; dim1[15:0]
    g1[3] = ((rows_remaining >> 16) & 0xFFFF) | (HD << 16); // dim1[31:16]; tile_dim0=64
    g1[4] = BM;                                          // tile_dim1=64; tile_dim2=0
    g1[5] = HD;                                          // tensor_dim0_stride[31:0]=64
    g1[6] = 0;                                           // stride0[47:32]; stride1[15:0]
    g1[7] = 0;
    i32x4 gz;
    gz[0] = 0; gz[1] = 0; gz[2] = 0; gz[3] = 0;          // groups 2/3 unused (2D)
    i32x8 gz8;
    gz8[0] = 0; gz8[1] = 0; gz8[2] = 0; gz8[3] = 0;
    gz8[4] = 0; gz8[5] = 0; gz8[6] = 0; gz8[7] = 0;
    __builtin_amdgcn_tensor_load_to_lds(g0, g1, gz, gz, gz8, 0);
}
#endif

__global__ __launch_bounds__(128, 1)
void retention_wmma_f32(const float* __restrict__ Qs, const float* __restrict__ Ks,
                        const float* __restrict__ V, float* __restrict__ O,
                        int seqlen, int prerot) {
    // [K0 | V0 | K1 | V1 | S-scratch(4 waves)]
    __shared__ __align__(16) float smem[4 * BUFSZ + NWAVE * 16 * LDSP];

    const int tid   = threadIdx.x;
    const int lane  = tid & 31;
    const int wave  = tid >> 5;
    const int ln16  = lane & 15;
    const int hi    = lane >> 4;           // half-wave: K-phase in fp32 A/B layouts
    const int b     = blockIdx.y;
    const int iBase = blockIdx.x * BM;

    const size_t bo = (size_t)b * (size_t)seqlen * HD;
    const float* Qb = Qs + bo;
    const float* Kb = Ks + bo;
    const float* Vb = V + bo;
    float*       Ob = O + bo;
    float*       sS = smem + 4 * BUFSZ + wave * (16 * LDSP);

#if HAVE_TDM
    // LDS aperture: generic address low 32 bits == LDS byte offset.
    const unsigned lds_base = (unsigned)(uintptr_t)(&smem[0]);
    const int use_tdm = prerot;
#else
    const int use_tdm = 0;
#endif

    // ---- Q A-fragments. fp32 A 16x4 layout: lane = row M (lane&15);
    // VGPR0 -> K = kb+2*hi, VGPR1 -> K = kb+2*hi+1.
    const int qrow = iBase + wave * 16 + ln16;
    v2f aQ[16];
    if (prerot) {
#pragma unroll
        for (int c = 0; c < 16; ++c) {
            int k0 = 4 * c + 2 * hi;
            aQ[c].x = Qb[(size_t)qrow * HD + k0];
            aQ[c].y = Qb[(size_t)qrow * HD + k0 + 1];
        }
    } else {
#pragma unroll
        for (int c = 0; c < 16; ++c) {
            int k0 = 4 * c + 2 * hi;
            float o0, o1;
            xpos_pair(Qb[(size_t)qrow * HD + k0], Qb[(size_t)qrow * HD + k0 + 1],
                      qrow, k0 >> 1, 1.0f, o0, o1);
            aQ[c].x = o0;
            aQ[c].y = o1;
        }
    }

    // Decay factors: gamma^diff = gamma^wbexp * gamma^r * gamma^(-16*nt)
    float gr[8], gnt[4];
#pragma unroll
    for (int r = 0; r < 8; ++r) gr[r] = exp2f((float)r * LOG2_GAMMA);
#pragma unroll
    for (int t = 0; t < 4; ++t) gnt[t] = exp2f((float)(-16 * t) * LOG2_GAMMA);

    v8f accO[4];
#pragma unroll
    for (int t = 0; t < 4; ++t) accO[t] = (v8f){0.f, 0.f, 0.f, 0.f, 0.f, 0.f, 0.f, 0.f};

    const int jEnd = blockIdx.x;                   // causal: key blocks <= query block

#if HAVE_TDM
    if (use_tdm) {                                 // prologue: DMA block 0 into buffer 0
        if (wave == 0) {
            tdm_load_tile(Kb, lds_base, seqlen);
            tdm_load_tile(Vb, lds_base + BUFSZ * 4u, seqlen);
            __builtin_amdgcn_s_wait_tensorcnt(0);
        }
        __syncthreads();
    }
#endif

    for (int j = 0; j <= jEnd; ++j) {
        const int jBase = j * BM;
        const int cur   = use_tdm ? (j & 1) : 0;
        const float* sK = smem + (2 * cur) * BUFSZ;
        const float* sV = smem + (2 * cur + 1) * BUFSZ;

        if (use_tdm) {
#if HAVE_TDM
            if (j < jEnd && wave == 0) {           // DMA next block into other buffer
                const int nb  = 1 - cur;
                const int njB = jBase + BM;
                tdm_load_tile(Kb + (size_t)njB * HD,
                              lds_base + (unsigned)(2 * nb) * BUFSZ * 4u, seqlen - njB);
                tdm_load_tile(Vb + (size_t)njB * HD,
                              lds_base + (unsigned)(2 * nb + 1) * BUFSZ * 4u,
                              seqlen - njB);
            }
#endif
        } else {
            __syncthreads();                       // protect buffer 0 from previous iter
            float* wK = smem;
            float* wV = smem + BUFSZ;
            if (prerot) {
                const float4* Ksrc = (const float4*)(Kb + (size_t)jBase * HD);
                const float4* Vsrc = (const float4*)(Vb + (size_t)jBase * HD);
                for (int idx = tid; idx < BM * (HD / 4); idx += 128) {
                    int row = idx >> 4;
                    int c4  = (idx & 15) * 4;
                    *(float4*)&wK[row * LDSP + c4] = Ksrc[idx];
                    *(float4*)&wV[row * LDSP + c4] = Vsrc[idx];
                }
            } else {
                for (int idx = tid; idx < BM * (HD / 2); idx += 128) {
                    int row  = idx >> 5;
                    int h    = idx & 31;
                    int kpos = jBase + row;
                    const float* kp = Kb + (size_t)kpos * HD + 2 * h;
                    float o0, o1;
                    xpos_pair(kp[0], kp[1], kpos, h, -1.0f, o0, o1);
                    wK[row * LDSP + 2 * h]     = o0;
                    wK[row * LDSP + 2 * h + 1] = o1;
                    const float* vp = Vb + (size_t)kpos * HD + 2 * h;
                    wV[row * LDSP + 2 * h]     = vp[0];
                    wV[row * LDSP + 2 * h + 1] = vp[1];
                }
            }
            if (j < jEnd) {                        // prefetch next block (16KB coverage)
                const char* nk = (const char*)(Kb + (size_t)(jBase + BM) * HD) + tid * 128;
                const char* nv = (const char*)(Vb + (size_t)(jBase + BM) * HD) + tid * 128;
                __builtin_prefetch(nk, 0, 1);
                __builtin_prefetch(nv, 0, 1);
            }
            __syncthreads();
        }

        // ---- S = Q K^T (wave's 16x64 strip); decay; spill to per-wave LDS scratch.
        const int wbexp = iBase + wave * 16 + 8 * hi - jBase - ln16;
        const float wb  = exp2f((float)wbexp * LOG2_GAMMA);
#pragma unroll
        for (int nt = 0; nt < 4; ++nt) {
            v8f acc = (v8f){0.f, 0.f, 0.f, 0.f, 0.f, 0.f, 0.f, 0.f};
#pragma unroll
            for (int c = 0; c < 16; ++c) {
                int k0 = 4 * c + 2 * hi;
                v2f bk;
                bk.x = sK[(nt * 16 + ln16) * LDSP + k0];
                bk.y = sK[(nt * 16 + ln16) * LDSP + k0 + 1];
                acc = __builtin_amdgcn_wmma_f32_16x16x4_f32(false, aQ[c], false, bk,
                                                            (short)0, acc, false, false);
            }
            const float wnt = wb * gnt[nt];
#pragma unroll
            for (int r = 0; r < 8; ++r) {
                int diff = wbexp + r - 16 * nt;                // qpos - kpos
                float w  = (diff >= 0) ? wnt * gr[r] : 0.0f;
                sS[(r + 8 * hi) * LDSP + nt * 16 + ln16] = acc[r] * w;
            }
        }
        // Same-wave LDS RAW (store S -> load S): DS ops are in-order per wave.

        // ---- O += S V.  A = S (16x64) in A layout; B = V columns from LDS.
#pragma unroll
        for (int vt = 0; vt < 4; ++vt) {
            v8f acc = accO[vt];
#pragma unroll
            for (int c = 0; c < 16; ++c) {
                int kb = 4 * c + 2 * hi;
                v2f a, bv;
                a.x  = sS[ln16 * LDSP + kb];
                a.y  = sS[ln16 * LDSP + kb + 1];
                bv.x = sV[kb * LDSP + vt * 16 + ln16];
                bv.y = sV[(kb + 1) * LDSP + vt * 16 + ln16];
                acc = __builtin_amdgcn_wmma_f32_16x16x4_f32(false, a, false, bv,
                                                            (short)0, acc, false, false);
            }
            accO[vt] = acc;
        }

#if HAVE_TDM
        if (use_tdm && j < jEnd) {
            if (wave == 0) __builtin_amdgcn_s_wait_tensorcnt(0);  // next buffers landed
            __syncthreads();  // everyone done reading cur; next buffer visible to all
        }
#endif
    }

    // ---- Store O (fully covers d_out).
#pragma unroll
    for (int vt = 0; vt < 4; ++vt) {
#pragma unroll
        for (int r = 0; r < 8; ++r) {
            int qp = iBase + wave * 16 + r + 8 * hi;
            Ob[(size_t)qp * HD + vt * 16 + ln16] = accO[vt][r];
        }
    }
}

extern "C" void kernel_launch(void* const* d_in, const int* in_sizes, int n_in,
                              void* d_out, int out_size, void* d_ws, size_t ws_size,
                              hipStream_t stream) {
    const float* Q = (const float*)d_in[0];
    const float* K = (const float*)d_in[1];
    const float* V = (const float*)d_in[2];
    float*       O = (float*)d_out;

    const int nQ = in_sizes[0];                    // B * S * HD
    const int Bn = nQ / (S_LEN * HD);

    const size_t need = 2ull * (size_t)nQ * sizeof(float);
    const int prerot  = (d_ws != nullptr && ws_size >= need) ? 1 : 0;

    const float* qsrc = Q;
    const float* ksrc = K;
    if (prerot) {
        float* Qh = (float*)d_ws;
        float* Kh = Qh + nQ;
        int total_pairs = nQ / 2;
        xpos_rotate<<<(total_pairs + 255) / 256, 256, 0, stream>>>(Q, K, Qh, Kh,
                                                                   total_pairs, S_LEN);
        qsrc = Qh;
        ksrc = Kh;
    }

    dim3 grid(S_LEN / BM, Bn);
    retention_wmma_f32<<<grid, 128, 0, stream>>>(qsrc, ksrc, V, O, S_LEN, prerot);
}